// KTModel_84275848282580
// MI455X (gfx1250) — compile-verified
//
#include <hip/hip_runtime.h>
#include <cstddef>

#define K_DIM 128
#define E_DIM 4096
#define M_DIM 512
#define NUM_UPDATE 3
#define THRESH 0.05f

#define YB 8            // e-direction block splits
#define WAVES 8         // waves per block in main kernel
#define TILES_TOTAL (E_DIM / 16)          // 256 tiles of 16 e's
#define TILES_PER_Y (TILES_TOTAL / YB)    // 32 -> 4 tiles per wave

typedef __attribute__((ext_vector_type(2))) float v2f;
typedef __attribute__((ext_vector_type(8))) float v8f;

// ---------------------------------------------------------------- init
__global__ void k_init(const float* __restrict__ U, const int* __restrict__ stu,
                       const int* __restrict__ kn, float* __restrict__ u_cur,
                       int* __restrict__ upd) {
  int tid = threadIdx.x;
  int s = stu[0];
  if (tid < K_DIM) {
    u_cur[tid] = U[(size_t)s * K_DIM + tid];
    upd[tid] = 0;
  }
  __syncthreads();
  if (tid < K_DIM) {
    int kk = kn[tid];
    if (kk >= 0 && kk < K_DIM) upd[kk] = 1;
  }
}

// ---------------------------------------------------------------- per-step pre
// su = q_kn @ u ; Ic, Zc scalars ; qdv = q_kn * dv
__global__ __launch_bounds__(256) void k_pre(
    const float* __restrict__ u, const float* __restrict__ q_kn,
    const float* __restrict__ d, const float* __restrict__ score,
    const float* __restrict__ gs, const float* __restrict__ A_emb,
    const float* __restrict__ gamma_c, const float* __restrict__ beta2,
    float* __restrict__ qdv, float* __restrict__ hIc, float* __restrict__ b2z) {
  __shared__ float u_s[K_DIM], dv_s[K_DIM];
  int tid = threadIdx.x;
  if (tid < K_DIM) {
    float uv = u[tid];
    u_s[tid] = uv;
    float dvv = uv - 0.5f;
    dv_s[tid] = (fabsf(dvv) > THRESH) ? dvv : 0.0f;
  }
  __syncthreads();
  int e = blockIdx.x * blockDim.x + tid;                    // E = 16*256 exactly
  const float* qr = q_kn + (size_t)e * K_DIM;
  float* qd = qdv + (size_t)e * K_DIM;
  float su = 0.f;
#pragma unroll 8
  for (int j = 0; j < K_DIM; ++j) {
    float q = qr[j];
    su += q * u_s[j];
    qd[j] = q * dv_s[j];
  }
  float de = d[e], sc = score[e];
  float t = sc - su / de;
  float yc = __expf(-t * t);
  float dot = A_emb[e * 3 + 0] * (1.f - gs[e * 2 + 0]) +
              A_emb[e * 3 + 1] * (1.f - gs[e * 2 + 1]) +
              A_emb[e * 3 + 2] * yc;
  float Ic = 1.f / (1.f + __expf(dot));                     // sigmoid(-dot)
  hIc[e] = 1.0f /*H*/ * Ic;
  float zc = 1.f / (1.f + __expf((gamma_c[e] / de) * (sc - 0.5f))) - 0.5f;
  b2z[e] = beta2[e] * zc;
}

// ---------------------------------------------------------------- per-step main
// Gsum[k,e] = sum_{j != k} B[k,e,j]*qdv[e,j]  via WMMA f32 16x16x4 ones-reduction.
// The single chunk containing j==k is peeled out of the hot loop so the
// streaming body is just: load B(b64,NT) + load qdv(b64) + v_pk_mul + v_wmma.
__global__ __launch_bounds__(256) void k_main(
    const float* __restrict__ B, const float* __restrict__ qdv,
    const float* __restrict__ hIc, const float* __restrict__ b2z,
    const float* __restrict__ W, const float* __restrict__ b1,
    const float* __restrict__ ukc, float* __restrict__ s_part) {
  const int k = blockIdx.x;
  const int yb = blockIdx.y;
  const int wave = threadIdx.x >> 5;
  const int lane = threadIdx.x & 31;
  const int m = lane & 15;        // A-matrix row (e within tile)
  const int half = lane >> 4;     // 0: K=j0..j0+1, 1: K=j0+2..j0+3 (f32 A layout)
  const size_t Bk = (size_t)k * ((size_t)E_DIM * K_DIM);
  const int kc = k & ~3;          // uniform: chunk holding the diagonal term
  // per-lane diagonal kill flags for the peeled chunk (j = kc + 2*half + c)
  const bool kill0 = (half == ((k >> 1) & 1)) && ((k & 1) == 0);
  const bool kill1 = (half == ((k >> 1) & 1)) && ((k & 1) == 1);

  v2f ones;
  ones[0] = 1.0f;
  ones[1] = 1.0f;
  float acc_s = 0.f;

  for (int t = yb * TILES_PER_Y + wave; t < (yb + 1) * TILES_PER_Y; t += WAVES) {
    const int e0 = t << 4;
    const float* bp = B + Bk + (size_t)(e0 + m) * K_DIM + 2 * half;
    const float* qp = qdv + (size_t)(e0 + m) * K_DIM + 2 * half;
    // warm the next tile's rows (one cacheline per lane)
    __builtin_prefetch(bp + (size_t)(WAVES * 16) * K_DIM, 0, 0);
    v8f acc = {0.f, 0.f, 0.f, 0.f, 0.f, 0.f, 0.f, 0.f};

    // plain chunks before the diagonal chunk
#pragma unroll 4
    for (int j0 = 0; j0 < kc; j0 += 4) {
      v2f bv = __builtin_nontemporal_load((const v2f*)(bp + j0));
      v2f qv = *(const v2f*)(qp + j0);
      v2f a = bv * qv;
      acc = __builtin_amdgcn_wmma_f32_16x16x4_f32(
          false, a, false, ones, (short)0, acc, false, false);
    }
    // the one chunk containing j==k: drop the diagonal term
    {
      v2f bv = __builtin_nontemporal_load((const v2f*)(bp + kc));
      v2f qv = *(const v2f*)(qp + kc);
      v2f a = bv * qv;
      a[0] = kill0 ? 0.f : a[0];
      a[1] = kill1 ? 0.f : a[1];
      acc = __builtin_amdgcn_wmma_f32_16x16x4_f32(
          false, a, false, ones, (short)0, acc, false, false);
    }
    // plain chunks after the diagonal chunk
#pragma unroll 4
    for (int j0 = kc + 4; j0 < K_DIM; j0 += 4) {
      v2f bv = __builtin_nontemporal_load((const v2f*)(bp + j0));
      v2f qv = *(const v2f*)(qp + j0);
      v2f a = bv * qv;
      acc = __builtin_amdgcn_wmma_f32_16x16x4_f32(
          false, a, false, ones, (short)0, acc, false, false);
    }

    // C/D layout: lanes 0-15, VGPR v -> M=v ; lanes 16-31, VGPR v -> M=8+v.
    // Active lanes 0..7 take e'=lane, lanes 16..23 take e'=lane-8.
    if ((lane & 8) == 0) {
      int idx = lane & 7;
      float g = acc[0];
      if (idx == 1) g = acc[1];
      if (idx == 2) g = acc[2];
      if (idx == 3) g = acc[3];
      if (idx == 4) g = acc[4];
      if (idx == 5) g = acc[5];
      if (idx == 6) g = acc[6];
      if (idx == 7) g = acc[7];
      int ep = ((lane & 16) >> 1) | idx;            // 0..15
      int e = e0 + ep;
      size_t ke = (size_t)k * E_DIM + e;
      float gkc = 1.f / (1.f + __expf(g)) - 1.f;    // sigmoid(-Gsum) - 1
      float wk = hIc[e] * (b1[ke] * gkc + b2z[e]);
      acc_s += ukc[ke] * W[ke] * wk;
    }
  }

  for (int o = 16; o > 0; o >>= 1) acc_s += __shfl_down(acc_s, o, 32);
  __shared__ float wred[WAVES];
  if (lane == 0) wred[wave] = acc_s;
  __syncthreads();
  if (threadIdx.x == 0) {
    float s = 0.f;
    for (int w = 0; w < WAVES; ++w) s += wred[w];
    s_part[k * YB + yb] = s;                        // deterministic: no atomics
  }
}

// ---------------------------------------------------------------- per-step post
__global__ void k_post(const float* __restrict__ s_part, const int* __restrict__ upd,
                       float* __restrict__ u_cur, float* __restrict__ u_prev,
                       float* __restrict__ diff_out) {
  __shared__ float red[K_DIM];
  int k = threadIdx.x;                              // blockDim = 128
  float s = 0.f;
  for (int y = 0; y < YB; ++y) s += s_part[k * YB + y];
  float old = u_cur[k];
  float un = upd[k] ? 1.f / (1.f + __expf(s)) : old;
  u_prev[k] = old;
  u_cur[k] = un;
  float dd = un - old;
  red[k] = dd * dd;
  __syncthreads();
  for (int o = K_DIM / 2; o > 0; o >>= 1) {
    if (k < o) red[k] += red[k + o];
    __syncthreads();
  }
  if (k == 0) *diff_out = sqrtf(red[0]);
}

// ---------------------------------------------------------------- final outputs
__global__ __launch_bounds__(512) void k_final(
    const float* __restrict__ u_cur, const float* __restrict__ u_prev,
    const float* __restrict__ q_kn, const float* __restrict__ d,
    const float* __restrict__ alpha, const float* __restrict__ gamma_e,
    const int* __restrict__ ex_id, float* __restrict__ out) {
  __shared__ float u_s[K_DIM];
  int tid = threadIdx.x;
  if (tid < K_DIM) u_s[tid] = u_cur[tid];
  __syncthreads();
  if (tid < K_DIM) {
    out[tid] = u_cur[tid];
    out[K_DIM + tid] = u_prev[tid];
  }
  if (tid < M_DIM) {
    int e = ex_id[tid];
    const float* qr = q_kn + (size_t)e * K_DIM;
    float su = 0.f;
#pragma unroll 8
    for (int j = 0; j < K_DIM; ++j) su += qr[j] * u_s[j];
    float uk = su / d[e] - 0.5f;
    out[2 * K_DIM + tid] = 1.f / (1.f + __expf(alpha[e] * uk + gamma_e[e]));
  }
}

// ---------------------------------------------------------------- launch
extern "C" void kernel_launch(void* const* d_in, const int* in_sizes, int n_in,
                              void* d_out, int out_size, void* d_ws, size_t ws_size,
                              hipStream_t stream) {
  const float* U     = (const float*)d_in[0];   // (S,K)
  const float* W     = (const float*)d_in[1];   // (K,E)
  const float* b1    = (const float*)d_in[2];   // (K,E)
  const float* b2    = (const float*)d_in[3];   // (E,1)
  const float* B     = (const float*)d_in[4];   // (K,E,K)
  const float* gs    = (const float*)d_in[5];   // (E,2)
  const float* A_emb = (const float*)d_in[6];   // (E,3)
  const float* gam_c = (const float*)d_in[7];   // (E,1)
  const float* gam_e = (const float*)d_in[8];   // (E,1)
  const float* alpha = (const float*)d_in[9];   // (E,1)
  const float* score = (const float*)d_in[10];  // (E,1)
  const float* ukc   = (const float*)d_in[11];  // (K,E)
  const float* q_kn  = (const float*)d_in[12];  // (E,K)
  const float* dvec  = (const float*)d_in[13];  // (E,1)
  const int*   stu   = (const int*)d_in[14];    // scalar
  const int*   kn    = (const int*)d_in[15];    // (K,)
  const int*   ex    = (const int*)d_in[16];    // (M,)
  float* out = (float*)d_out;

  float* ws     = (float*)d_ws;
  float* u_cur  = ws;                        // 128
  float* u_prev = ws + 128;                  // 128
  float* s_part = ws + 256;                  // K*YB = 1024
  int*   upd    = (int*)(ws + 1280);         // 128
  float* hIc    = ws + 1408;                 // 4096
  float* b2z    = ws + 1408 + E_DIM;         // 4096
  float* qdv    = ws + 1408 + 2 * E_DIM;     // E*K = 524288  (~2.1MB total)

  k_init<<<1, 128, 0, stream>>>(U, stu, kn, u_cur, upd);
  for (int step = 0; step < NUM_UPDATE; ++step) {
    k_pre<<<E_DIM / 256, 256, 0, stream>>>(u_cur, q_kn, dvec, score, gs, A_emb,
                                           gam_c, b2, qdv, hIc, b2z);
    dim3 g(K_DIM, YB);
    k_main<<<g, 256, 0, stream>>>(B, qdv, hIc, b2z, W, b1, ukc, s_part);
    k_post<<<1, 128, 0, stream>>>(s_part, upd, u_cur, u_prev,
                                  out + 2 * K_DIM + M_DIM + step);
  }
  k_final<<<1, 512, 0, stream>>>(u_cur, u_prev, q_kn, dvec, alpha, gam_e, ex, out);
}